// GraphEmbedder_40587440947286
// MI455X (gfx1250) — compile-verified
//
#include <hip/hip_runtime.h>

typedef float v2f __attribute__((ext_vector_type(2)));
typedef float v8f __attribute__((ext_vector_type(8)));

#define N_OUT_DIM 128   // both layers emit 128 features

// ---------------- utility kernels ----------------

__global__ void zero_kernel(float* __restrict__ p, long long n) {
    long long i = (long long)blockIdx.x * blockDim.x + threadIdx.x;
    if (i < n) p[i] = 0.0f;
}

__global__ void degree_kernel(const long long* __restrict__ dst,
                              float* __restrict__ deg, int n_edges) {
    int e = blockIdx.x * blockDim.x + threadIdx.x;
    if (e < n_edges) atomicAdd(&deg[(int)dst[e]], 1.0f);
}

__global__ void invdeg_kernel(float* __restrict__ deg, int n_nodes) {
    int i = blockIdx.x * blockDim.x + threadIdx.x;
    if (i < n_nodes) {
        float d = deg[i];
        deg[i] = 1.0f / fmaxf(d, 1.0f);   // matches reference max(deg,1)
    }
}

// ---------------- edge scatter (L2-resident atomics) ----------------

template <int D>
__global__ void scatter_add_kernel(const float* __restrict__ feat,
                                   const long long* __restrict__ src,
                                   const long long* __restrict__ dst,
                                   float* __restrict__ agg, int n_edges) {
    constexpr int CH = D / 4;                       // float4 chunks per edge
    long long t = (long long)blockIdx.x * blockDim.x + threadIdx.x;
    long long total = (long long)n_edges * CH;
    if (t >= total) return;
    int e = (int)(t / CH);
    int c = (int)(t % CH);
    int s = (int)src[e];
    int d = (int)dst[e];
    const float4 v = *(const float4*)(feat + (size_t)s * D + c * 4);
    float* o = agg + (size_t)d * D + c * 4;
    atomicAdd(o + 0, v.x);
    atomicAdd(o + 1, v.y);
    atomicAdd(o + 2, v.z);
    atomicAdd(o + 3, v.w);
}

// ---------------- fused SAGE GEMM: out = (agg*invdeg)@Wl.T + x@Wr.T + b ----------------
// One wave -> one 16x16 tile of the [n_nodes x 128] output.
// FP32 WMMA (V_WMMA_F32_16X16X4_F32) keeps exact fp32 parity with the reference.

template <int K, bool RELU>
__global__ void sage_gemm_kernel(const float* __restrict__ agg,     // [N,K] (unnormalized sums)
                                 const float* __restrict__ invdeg,  // [N]
                                 const float* __restrict__ xin,     // [N,K]
                                 const float* __restrict__ Wl,      // [128,K] row-major
                                 const float* __restrict__ Wr,      // [128,K] row-major
                                 const float* __restrict__ bias,    // [128]
                                 float* __restrict__ out,           // [N,128]
                                 int n_nodes) {
    const int lane  = threadIdx.x & 31;
    const int wave  = threadIdx.x >> 5;            // 4 waves / block
    const int tile  = blockIdx.x * 4 + wave;
    const int n_mtiles = (n_nodes + 15) >> 4;
    const int mtile = tile >> 3;                   // 128/16 = 8 N-tiles
    const int ntile = tile & 7;
    if (mtile >= n_mtiles) return;                 // wave-uniform; EXEC stays all-ones

    const int idx   = lane & 15;                   // M for A-frag, N for B-frag
    const int khalf = lane >> 4;                   // selects K pair {0,1} vs {2,3}

    int row = mtile * 16 + idx;
    if (row >= n_nodes) row = n_nodes - 1;         // clamp (tiles are exact here anyway)
    const int ncol = ntile * 16 + idx;

    const float idg = invdeg[row];

    const float* aRow  = agg + (size_t)row  * K + 2 * khalf;   // A: lane=(k/2)*16+M, vgpr=k%2
    const float* xRow  = xin + (size_t)row  * K + 2 * khalf;
    const float* wlRow = Wl  + (size_t)ncol * K + 2 * khalf;   // B: lane=(k/2)*16+N, vgpr=k%2
    const float* wrRow = Wr  + (size_t)ncol * K + 2 * khalf;

    v8f c = {0.f, 0.f, 0.f, 0.f, 0.f, 0.f, 0.f, 0.f};

    // mean @ Wl.T   (mean-normalization folded into the A-fragment load)
#pragma unroll
    for (int k = 0; k < K; k += 4) {
        v2f a = *(const v2f*)(aRow + k);
        a *= idg;
        v2f b = *(const v2f*)(wlRow + k);
        c = __builtin_amdgcn_wmma_f32_16x16x4_f32(false, a, false, b,
                                                  (short)0, c, false, false);
    }
    // x @ Wr.T
#pragma unroll
    for (int k = 0; k < K; k += 4) {
        v2f a = *(const v2f*)(xRow + k);
        v2f b = *(const v2f*)(wrRow + k);
        c = __builtin_amdgcn_wmma_f32_16x16x4_f32(false, a, false, b,
                                                  (short)0, c, false, false);
    }

    // Epilogue. C/D layout: vgpr r holds (M = r + 8*khalf, N = lane&15).
    const float bv    = bias[ncol];
    const int   mBase = mtile * 16 + 8 * khalf;
#pragma unroll
    for (int r = 0; r < 8; ++r) {
        const int mr = mBase + r;
        if (mr < n_nodes) {
            float v = c[r] + bv;
            if (RELU) v = fmaxf(v, 0.0f);
            out[(size_t)mr * N_OUT_DIM + ncol] = v;
        }
    }
}

// ---------------- launch ----------------

extern "C" void kernel_launch(void* const* d_in, const int* in_sizes, int n_in,
                              void* d_out, int out_size, void* d_ws, size_t ws_size,
                              hipStream_t stream) {
    const float*     x    = (const float*)d_in[0];
    const long long* ei   = (const long long*)d_in[1];   // int64 [2, E]
    const float*     W1l  = (const float*)d_in[2];       // [128,64]
    const float*     W1r  = (const float*)d_in[3];       // [128,64]
    const float*     b1   = (const float*)d_in[4];       // [128]
    const float*     W2l  = (const float*)d_in[5];       // [128,128]
    const float*     W2r  = (const float*)d_in[6];       // [128,128]
    const float*     b2   = (const float*)d_in[7];       // [128]
    float*           outp = (float*)d_out;

    const int n_nodes = in_sizes[0] / 64;   // 50000
    const int n_edges = in_sizes[1] / 2;    // 800000
    const long long* src = ei;
    const long long* dst = ei + n_edges;

    // workspace layout: deg/invdeg | agg1[N,64] | agg2[N,128] | h[N,128]
    float* deg  = (float*)d_ws;
    float* agg1 = deg  + (size_t)n_nodes;
    float* agg2 = agg1 + (size_t)n_nodes * 64;
    float* h    = agg2 + (size_t)n_nodes * 128;

    // zero deg + agg1 + agg2 in one contiguous pass (h is fully overwritten)
    const long long zn = (long long)n_nodes * (1 + 64 + 128);
    zero_kernel<<<(int)((zn + 255) / 256), 256, 0, stream>>>(deg, zn);

    // degrees -> inverse degrees (shared by both layers)
    degree_kernel<<<(n_edges + 255) / 256, 256, 0, stream>>>(dst, deg, n_edges);
    invdeg_kernel<<<(n_nodes + 255) / 256, 256, 0, stream>>>(deg, n_nodes);

    // layer 1: scatter x into agg1, fused mean+GEMM+bias+ReLU -> h
    {
        long long tot = (long long)n_edges * (64 / 4);
        scatter_add_kernel<64><<<(int)((tot + 255) / 256), 256, 0, stream>>>(
            x, src, dst, agg1, n_edges);
        const int tiles  = ((n_nodes + 15) / 16) * 8;
        const int blocks = (tiles + 3) / 4;
        sage_gemm_kernel<64, true><<<blocks, 128, 0, stream>>>(
            agg1, deg, x, W1l, W1r, b1, h, n_nodes);
    }

    // layer 2: scatter h into agg2, fused mean+GEMM+bias -> d_out
    {
        long long tot = (long long)n_edges * (128 / 4);
        scatter_add_kernel<128><<<(int)((tot + 255) / 256), 256, 0, stream>>>(
            h, src, dst, agg2, n_edges);
        const int tiles  = ((n_nodes + 15) / 16) * 8;
        const int blocks = (tiles + 3) / 4;
        sage_gemm_kernel<128, false><<<blocks, 128, 0, stream>>>(
            agg2, deg, h, W2l, W2r, b2, outp, n_nodes);
    }
}